// MinCut_24266565222650
// MI455X (gfx1250) — compile-verified
//
#include <hip/hip_runtime.h>
#include <hip/hip_bf16.h>
#include <math.h>

// ---------------------------------------------------------------------------
// MinCut GCN pool for MI455X (gfx1250, wave32).
// GEMMs via V_WMMA_F32_16X16X4_F32 (fp32 in/out, matrix pipe), B staged in LDS
// with zero padding so the inner loop has no divergent predication.
// Edge propagation via GLOBAL_ATOMIC_ADD_F32 (memory-bound part, ~200MB total).
// ---------------------------------------------------------------------------

typedef __attribute__((ext_vector_type(2))) float v2f;
typedef __attribute__((ext_vector_type(8))) float v8f;

#define IN_CH 128
#define HID   32
#define KCL   10
#define CCL   7

// ---------------- zero init of atomic accumulators -------------------------
__global__ void zero_kernel(float* __restrict__ buf, int n) {
    int i = blockIdx.x * blockDim.x + threadIdx.x;
    if (i < n) buf[i] = 0.0f;
}

// ---------------- degree count (col for norm, row for mincut_den) ----------
__global__ void degree_kernel(const int* __restrict__ row, const int* __restrict__ col,
                              float* __restrict__ deg_col, float* __restrict__ deg_row, int E) {
    int e = blockIdx.x * blockDim.x + threadIdx.x;
    if (e < E) {
        atomicAdd(&deg_col[col[e]], 1.0f);
        atomicAdd(&deg_row[row[e]], 1.0f);
    }
}

__global__ void dis_kernel(const float* __restrict__ deg_col, float* __restrict__ dis, int n) {
    int i = blockIdx.x * blockDim.x + threadIdx.x;
    if (i < n) dis[i] = rsqrtf(deg_col[i] + 1.0f);
}

// ---------------- WMMA f32 GEMM: Out[M,Nmat] = A[M,Kd] @ B[Kd,Nmat] (+bias) -
// REQUIRES: M % 16 == 0 (N=12000 -> 750 tiles), Kd % 4 == 0, Kd <= 128,
//           Nmat <= NTILES*16.
// One wave computes a 16 x (NTILES*16) strip: A loaded once per k-step (b64),
// reused across NTILES WMMAs. B staged in LDS zero-padded to PADN columns.
// A 16x4 layout (ISA 7.12.2): lanes 0-15 hold K={0,1}; lanes 16-31 K={2,3}.
// B 4x16 row-striped across lanes: v0 = rows K=0/K=2, v1 = rows K=1/K=3.
// C/D: VGPR r -> M=r (lanes 0-15, N=lane) / M=r+8 (lanes 16-31, N=lane-16).
template <bool ADD_BIAS, int NTILES>
__global__ void gemm_wmma_kernel(const float* __restrict__ A, const float* __restrict__ B,
                                 const float* __restrict__ bias, float* __restrict__ Out,
                                 int M, int Kd, int Nmat) {
    constexpr int PADN = NTILES * 16;
    __shared__ float Bs[128 * PADN];

    int tid = threadIdx.x;
    // cooperative stage of B (zero-padded in N)
    for (int idx = tid; idx < Kd * PADN; idx += blockDim.x) {
        int k = idx / PADN, n = idx - k * PADN;
        Bs[idx] = (n < Nmat) ? B[k * Nmat + n] : 0.0f;
    }
    __syncthreads();

    int wave = blockIdx.x * (blockDim.x >> 5) + (tid >> 5);
    int tilesM = M >> 4;
    if (wave >= tilesM) return;            // wave-uniform: EXEC all-1s at WMMA
    int m0 = wave << 4;

    int lane = tid & 31;
    int half = lane >> 4;                  // 0: lanes 0-15, 1: lanes 16-31
    int mr   = lane & 15;

    const float* Arow = A + (size_t)(m0 + mr) * Kd;

    v8f c[NTILES];
#pragma unroll
    for (int t = 0; t < NTILES; ++t) c[t] = (v8f){0.f,0.f,0.f,0.f,0.f,0.f,0.f,0.f};

    for (int kk = 0; kk < Kd; kk += 4) {
        int ka = kk + 2 * half;
        v2f a = *(const v2f*)(Arow + ka);          // global_load_b64, unpredicated
#pragma unroll
        for (int t = 0; t < NTILES; ++t) {
            v2f b;
            b[0] = Bs[ka * PADN + t * 16 + mr];    // ds_load_b32
            b[1] = Bs[(ka + 1) * PADN + t * 16 + mr];
            c[t] = __builtin_amdgcn_wmma_f32_16x16x4_f32(false, a, false, b,
                                                         (short)0, c[t], false, false);
        }
    }

#pragma unroll
    for (int t = 0; t < NTILES; ++t) {
        int ncol = t * 16 + mr;
        if (ncol < Nmat) {
            float bs = ADD_BIAS ? bias[ncol] : 0.0f;
#pragma unroll
            for (int r = 0; r < 8; ++r) {
                int orow = m0 + r + 8 * half;
                Out[(size_t)orow * Nmat + ncol] = c[t][r] + bs;
            }
        }
    }
}

// ---------------- GCN propagation ------------------------------------------
// out[i,:] = h[i,:] * dis[i]^2   (self-loop term; also initializes buffer)
__global__ void self_loop_kernel(const float* __restrict__ h, const float* __restrict__ dis,
                                 float* __restrict__ out, int total /* = N*HID */) {
    int idx = blockIdx.x * blockDim.x + threadIdx.x;
    if (idx < total) {
        int i = idx >> 5;  // HID == 32
        float d = dis[i];
        out[idx] = h[idx] * d * d;
    }
}

// out[col,:] += h[row,:] * dis[row]*dis[col]  — one wave (32 lanes) per edge
__global__ void edge_scatter_kernel(const float* __restrict__ h, const float* __restrict__ dis,
                                    const int* __restrict__ row, const int* __restrict__ col,
                                    float* __restrict__ out, int E) {
    int idx = blockIdx.x * blockDim.x + threadIdx.x;
    int e = idx >> 5, ch = idx & 31;
    if (e < E) {
        int r = row[e], c = col[e];
        float v = h[r * HID + ch] * dis[r] * dis[c];
        atomicAdd(&out[c * HID + ch], v);   // global_atomic_add_f32, no return
    }
}

__global__ void bias_relu_kernel(float* __restrict__ out, const float* __restrict__ bias,
                                 int total) {
    int idx = blockIdx.x * blockDim.x + threadIdx.x;
    if (idx < total) {
        int ch = idx & 31;
        out[idx] = fmaxf(out[idx] + bias[ch], 0.0f);
    }
}

// ---------------- softmax over K=10 clusters -------------------------------
__global__ void softmax10_kernel(const float* __restrict__ p, float* __restrict__ s, int n) {
    int i = blockIdx.x * blockDim.x + threadIdx.x;
    if (i >= n) return;
    float m = -3.0e38f;
    for (int k = 0; k < KCL; ++k) m = fmaxf(m, p[i * KCL + k]);
    float e[KCL], sum = 0.0f;
    for (int k = 0; k < KCL; ++k) { e[k] = expf(p[i * KCL + k] - m); sum += e[k]; }
    float inv = 1.0f / sum;
    for (int k = 0; k < KCL; ++k) s[i * KCL + k] = e[k] * inv;
}

// ---------------- log_softmax over C=7 classes -> d_out --------------------
__global__ void logsoftmax7_kernel(const float* __restrict__ lg, float* __restrict__ out, int n) {
    int i = blockIdx.x * blockDim.x + threadIdx.x;
    if (i >= n) return;
    float m = -3.0e38f;
    for (int k = 0; k < CCL; ++k) m = fmaxf(m, lg[i * CCL + k]);
    float sum = 0.0f;
    for (int k = 0; k < CCL; ++k) sum += expf(lg[i * CCL + k] - m);
    float lse = m + logf(sum);
    for (int k = 0; k < CCL; ++k) out[i * CCL + k] = lg[i * CCL + k] - lse;
}

// ---------------- mincut numerator: sum_e  s[row_e] . s[col_e] -------------
__global__ void mincut_num_kernel(const float* __restrict__ s, const int* __restrict__ row,
                                  const int* __restrict__ col, float* __restrict__ scal, int E) {
    int e = blockIdx.x * blockDim.x + threadIdx.x;
    float v = 0.0f;
    if (e < E) {
        int r = row[e], c = col[e];
        for (int k = 0; k < KCL; ++k) v += s[r * KCL + k] * s[c * KCL + k];
    }
    for (int off = 16; off > 0; off >>= 1) v += __shfl_down(v, off, 32);
    if ((threadIdx.x & 31) == 0) atomicAdd(&scal[0], v);
}

// ---------------- mincut denominator: sum_i d[i] * ||s_i||^2 ---------------
__global__ void mincut_den_kernel(const float* __restrict__ s, const float* __restrict__ deg_row,
                                  float* __restrict__ scal, int n) {
    int i = blockIdx.x * blockDim.x + threadIdx.x;
    float v = 0.0f;
    if (i < n) {
        float ss = 0.0f;
        for (int k = 0; k < KCL; ++k) { float t = s[i * KCL + k]; ss += t * t; }
        v = deg_row[i] * ss;
    }
    for (int off = 16; off > 0; off >>= 1) v += __shfl_down(v, off, 32);
    if ((threadIdx.x & 31) == 0) atomicAdd(&scal[1], v);
}

// ---------------- S^T S  (10x10), LDS-staged -------------------------------
#define SS_CHUNK 64
__global__ void sts_kernel(const float* __restrict__ s, float* __restrict__ ssAcc, int n) {
    __shared__ float sh[SS_CHUNK * KCL];
    int tid = threadIdx.x;
    bool act = tid < KCL * KCL;
    int i = tid / KCL, j = tid % KCL;
    float acc = 0.0f;
    for (int base = blockIdx.x * SS_CHUNK; base < n; base += gridDim.x * SS_CHUNK) {
        int cnt = min(SS_CHUNK, n - base);
        for (int t = tid; t < cnt * KCL; t += blockDim.x) sh[t] = s[base * KCL + t];
        __syncthreads();
        if (act)
            for (int r = 0; r < cnt; ++r) acc += sh[r * KCL + i] * sh[r * KCL + j];
        __syncthreads();
    }
    if (act) atomicAdd(&ssAcc[tid], acc);
}

// ---------------- final scalars: mc_loss, o_loss ---------------------------
__global__ void finish_kernel(const float* __restrict__ scal, const float* __restrict__ ssAcc,
                              float* __restrict__ out2) {
    if (blockIdx.x == 0 && threadIdx.x == 0) {
        out2[0] = -(scal[0] / scal[1]);
        float fr = 0.0f;
        for (int t = 0; t < KCL * KCL; ++t) fr += ssAcc[t] * ssAcc[t];
        fr = sqrtf(fr);
        float invfr = 1.0f / fr;
        float diag = rsqrtf((float)KCL);
        float o = 0.0f;
        for (int i = 0; i < KCL; ++i)
            for (int j = 0; j < KCL; ++j) {
                float v = ssAcc[i * KCL + j] * invfr - ((i == j) ? diag : 0.0f);
                o += v * v;
            }
        out2[1] = sqrtf(o);
    }
}

// ===========================================================================
extern "C" void kernel_launch(void* const* d_in, const int* in_sizes, int n_in,
                              void* d_out, int out_size, void* d_ws, size_t ws_size,
                              hipStream_t stream) {
    const float* x   = (const float*)d_in[0];
    const int*   ei  = (const int*)d_in[1];
    const float* W1  = (const float*)d_in[2];
    const float* b1  = (const float*)d_in[3];
    const float* W2  = (const float*)d_in[4];
    const float* b2  = (const float*)d_in[5];
    const float* Wp  = (const float*)d_in[6];
    const float* bp  = (const float*)d_in[7];
    const float* Wc  = (const float*)d_in[8];
    const float* bc  = (const float*)d_in[9];
    float* outF = (float*)d_out;

    const int N = in_sizes[0] / IN_CH;     // 12000 (divisible by 16)
    const int E = in_sizes[1] / 2;         // 384000
    const int* erow = ei;
    const int* ecol = ei + E;

    // workspace layout (floats)
    float* ws      = (float*)d_ws;
    float* deg_col = ws;                   // N
    float* deg_row = ws + N;               // N
    float* scal    = ws + 2 * N;           // 2
    float* ssAcc   = ws + 2 * N + 2;       // 100
    float* dis     = ws + 2 * N + 102;     // N
    float* hT      = dis + N;              // N*32 (XW scratch)
    float* h1      = hT  + (size_t)N * HID;// N*32
    float* h2      = h1  + (size_t)N * HID;// N*32
    float* p       = h2  + (size_t)N * HID;// N*10 (reused for C=7 logits)
    float* s       = p   + (size_t)N * KCL;// N*10

    const int T = 256;
    auto cdiv = [](int a, int b) { return (a + b - 1) / b; };
    const int tilesM  = N / 16;            // 750 waves per GEMM
    const int gemmBlk = cdiv(tilesM, 4);   // 128 threads = 4 waves/block

    // 1) zero atomic accumulators (deg_col, deg_row, scal, ssAcc)
    int zn = 2 * N + 102;
    zero_kernel<<<cdiv(zn, T), T, 0, stream>>>(ws, zn);

    // 2) degrees + normalization
    degree_kernel<<<cdiv(E, T), T, 0, stream>>>(erow, ecol, deg_col, deg_row, E);
    dis_kernel<<<cdiv(N, T), T, 0, stream>>>(deg_col, dis, N);

    // 3) conv1: hT = x@W1 ; propagate ; +b1 ; relu -> h1
    gemm_wmma_kernel<false, 2><<<gemmBlk, 128, 0, stream>>>(x, W1, nullptr, hT, N, IN_CH, HID);
    self_loop_kernel<<<cdiv(N * HID, T), T, 0, stream>>>(hT, dis, h1, N * HID);
    edge_scatter_kernel<<<cdiv(E * 32, T), T, 0, stream>>>(hT, dis, erow, ecol, h1, E);
    bias_relu_kernel<<<cdiv(N * HID, T), T, 0, stream>>>(h1, b1, N * HID);

    // 4) conv2: hT = h1@W2 ; propagate ; +b2 ; relu -> h2
    gemm_wmma_kernel<false, 2><<<gemmBlk, 128, 0, stream>>>(h1, W2, nullptr, hT, N, HID, HID);
    self_loop_kernel<<<cdiv(N * HID, T), T, 0, stream>>>(hT, dis, h2, N * HID);
    edge_scatter_kernel<<<cdiv(E * 32, T), T, 0, stream>>>(hT, dis, erow, ecol, h2, E);
    bias_relu_kernel<<<cdiv(N * HID, T), T, 0, stream>>>(h2, b2, N * HID);

    // 5) cluster assignment: s = softmax(h2@Wp + bp)
    gemm_wmma_kernel<true, 1><<<gemmBlk, 128, 0, stream>>>(h2, Wp, bp, p, N, HID, KCL);
    softmax10_kernel<<<cdiv(N, T), T, 0, stream>>>(p, s, N);

    // 6) classifier: log_softmax(h2@Wc + bc) -> d_out[0 .. N*C)
    gemm_wmma_kernel<true, 1><<<gemmBlk, 128, 0, stream>>>(h2, Wc, bc, p, N, HID, CCL);
    logsoftmax7_kernel<<<cdiv(N, T), T, 0, stream>>>(p, outF, N);

    // 7) losses
    mincut_num_kernel<<<cdiv(E, T), T, 0, stream>>>(s, erow, ecol, scal, E);
    mincut_den_kernel<<<cdiv(N, T), T, 0, stream>>>(s, deg_row, scal, N);
    sts_kernel<<<30, 128, 0, stream>>>(s, ssAcc, N);
    finish_kernel<<<1, 32, 0, stream>>>(scal, ssAcc, outF + (size_t)N * CCL);
}